// Ewald_38173669327380
// MI455X (gfx1250) — compile-verified
//
#include <hip/hip_runtime.h>
#include <hip/hip_bf16.h>

// ---------------------------------------------------------------------------
// Ewald real-space multipole kernel for MI455X (gfx1250, wave32).
//
//  K1 ewald_multipole : VALU, all dipole/quadrupole source terms (pair-coupled
//                       operator -> structured per-pair contraction, ~100 flop
//                       per pair-channel after analytic reduction).
//  K2 ewald_charge    : WMMA f32 16x16x4. phi_qq = (nc*g) @ q and
//                       E_charge_c = (-nc*D1*d_c) @ q are true [512x512]@[512x8]
//                       GEMMs -> native f32 WMMA path.
//  K3 ewald_finalize  : assembles induced charges/dipoles + deterministic
//                       LDS tree reduction of the potential.
// ---------------------------------------------------------------------------

typedef float v2f __attribute__((ext_vector_type(2)));
typedef float v8f __attribute__((ext_vector_type(8)));

#define NPART 512
#define NQC   8
#define NJQ   (NPART * NQC)

#define A_ERF 0.70710678118654752f               /* a = 1/(sigma*sqrt(2)), sigma=1 */
#define A2C   0.5f                               /* a^2 */
#define GPREF 0.79788456080286536f               /* 2a/sqrt(pi) */
#define NCC   (90.4756f / 6.28318530717958648f)  /* norm_factor / (2*pi) */

// workspace layout: W_COUNT arrays of NJQ floats each (13*4096*4 = 208 KB)
enum {
  W_PU = 0, W_PQ, W_EUX, W_EUY, W_EUZ, W_EQX, W_EQY, W_EQZ, W_SQQ,
  W_PG, W_ECX, W_ECY, W_ECZ, W_COUNT
};

__device__ __forceinline__ void radial_all(float r2, float& D1, float& D2,
                                           float& D3, float& D4) {
  float rr  = sqrtf(r2);
  float ir  = 1.0f / rr;
  float ir2 = ir * ir, ir3 = ir2 * ir, ir4 = ir2 * ir2, ir5 = ir4 * ir;
  float ir6 = ir4 * ir2, ir7 = ir6 * ir, ir8 = ir4 * ir4, ir9 = ir8 * ir;
  float E = erff(A_ERF * rr);
  float G = GPREF * expf(-A2C * r2);
  D1 = G * ir2 - E * ir3;
  D2 = 3.f * E * ir5 - 3.f * G * ir4 - (2.f * A2C) * G * ir2;
  D3 = -15.f * E * ir7 + 15.f * G * ir6 + (10.f * A2C) * G * ir4
       + (4.f * A2C * A2C) * G * ir2;
  D4 = 105.f * E * ir9 - 105.f * G * ir8 - (70.f * A2C) * G * ir6
       - (28.f * A2C * A2C) * G * ir4 - (8.f * A2C * A2C * A2C) * G * ir2;
}

// ---------------------------------------------------------------------------
// Kernel 1: dipole & quadrupole source terms (VALU). One thread per (j, q).
// ---------------------------------------------------------------------------
__global__ __launch_bounds__(256) void ewald_multipole(
    const float* __restrict__ r, const float* __restrict__ u,
    const float* __restrict__ quad, float* __restrict__ ws) {
  __shared__ float s_r[NPART * 3];
  const int tid = threadIdx.x;
  for (int t = tid; t < NPART * 3; t += 256) s_r[t] = r[t];
  __syncthreads();

  const int gid = blockIdx.x * 256 + tid;   // 0..4095
  const int j   = gid >> 3;
  const int qc  = gid & 7;

  const float rjx = s_r[3 * j + 0];
  const float rjy = s_r[3 * j + 1];
  const float rjz = s_r[3 * j + 2];

  // target quadrupole -> S_j = Q_j + Q_j^T, c_j = tr(Q_j)
  const float* Qj = quad + (j * NQC + qc) * 9;
  const float a00 = Qj[0], a01 = Qj[1], a02 = Qj[2];
  const float a10 = Qj[3], a11 = Qj[4], a12 = Qj[5];
  const float a20 = Qj[6], a21 = Qj[7], a22 = Qj[8];
  const float cj   = a00 + a11 + a22;
  const float Sjxx = 2.f * a00, Sjyy = 2.f * a11, Sjzz = 2.f * a22;
  const float Sjxy = a01 + a10, Sjxz = a02 + a20, Sjyz = a12 + a21;

  float pu = 0.f, pQ = 0.f;
  float Eux = 0.f, Euy = 0.f, Euz = 0.f;
  float EQx = 0.f, EQy = 0.f, EQz = 0.f;
  float sQ = 0.f;

  for (int i = 0; i < NPART; ++i) {
    if (i + 16 < NPART) {  // gfx1250 global_prefetch_b8 of upcoming rows
      __builtin_prefetch(quad + ((i + 16) * NQC + qc) * 9, 0, 3);
      __builtin_prefetch(u + ((i + 16) * NQC + qc) * 3, 0, 3);
    }
    if (i == j) continue;  // off-diagonal only (self term removed)

    const float dx = rjx - s_r[3 * i + 0];
    const float dy = rjy - s_r[3 * i + 1];
    const float dz = rjz - s_r[3 * i + 2];
    const float r2 = dx * dx + dy * dy + dz * dz;
    float D1, D2, D3, D4;
    radial_all(r2, D1, D2, D3, D4);

    const float* up = u + (i * NQC + qc) * 3;
    const float ux = up[0], uy = up[1], uz = up[2];
    const float* Qp = quad + (i * NQC + qc) * 9;
    const float b00 = Qp[0], b01 = Qp[1], b02 = Qp[2];
    const float b10 = Qp[3], b11 = Qp[4], b12 = Qp[5];
    const float b20 = Qp[6], b21 = Qp[7], b22 = Qp[8];
    const float ci  = b00 + b11 + b22;
    const float sxx = 2.f * b00, syy = 2.f * b11, szz = 2.f * b22;
    const float sxy = b01 + b10, sxz = b02 + b20, syz = b12 + b21;

    const float bb = ux * dx + uy * dy + uz * dz;          // u_i . d
    const float wix = sxx * dx + sxy * dy + sxz * dz;      // (S_i d)
    const float wiy = sxy * dx + syy * dy + syz * dz;
    const float wiz = sxz * dx + syz * dy + szz * dz;
    const float ei = 0.5f * (wix * dx + wiy * dy + wiz * dz);  // d.Q_i.d
    const float wjx = Sjxx * dx + Sjxy * dy + Sjxz * dz;       // (S_j d)
    const float wjy = Sjxy * dx + Sjyy * dy + Sjyz * dz;
    const float wjz = Sjxz * dx + Sjyz * dy + Sjzz * dz;
    const float ej = 0.5f * (wjx * dx + wjy * dy + wjz * dz);  // d.Q_j.d
    const float ww = wix * wjx + wiy * wjy + wiz * wjz;
    const float SS = sxx * Sjxx + syy * Sjyy + szz * Sjzz
                     + 2.f * (sxy * Sjxy + sxz * Sjxz + syz * Sjyz);

    pu += D1 * bb;
    pQ += D2 * ei + D1 * ci;
    Eux += D2 * bb * dx + D1 * ux;
    Euy += D2 * bb * dy + D1 * uy;
    Euz += D2 * bb * dz + D1 * uz;
    EQx += D3 * ei * dx + D2 * (ci * dx + wix);
    EQy += D3 * ei * dy + D2 * (ci * dy + wiy);
    EQz += D3 * ei * dz + D2 * (ci * dz + wiz);
    sQ += D4 * ei * ej + D3 * (cj * ei + ej * ci + ww)
          + D2 * (ci * cj + 0.5f * SS);
  }

  const int idx = gid;
  ws[W_PU * NJQ + idx]  = -NCC * pu;          // phi_u
  ws[W_PQ * NJQ + idx]  = 0.5f * NCC * pQ;    // phi_Q
  ws[W_EUX * NJQ + idx] = NCC * Eux;          // E_u
  ws[W_EUY * NJQ + idx] = NCC * Euy;
  ws[W_EUZ * NJQ + idx] = NCC * Euz;
  ws[W_EQX * NJQ + idx] = -0.5f * NCC * EQx;  // E_Q
  ws[W_EQY * NJQ + idx] = -0.5f * NCC * EQy;
  ws[W_EQZ * NJQ + idx] = -0.5f * NCC * EQz;
  ws[W_SQQ * NJQ + idx] = 0.125f * NCC * sQ;  // 0.125 * Q_j : QQ_j (pot term)
}

// ---------------------------------------------------------------------------
// Kernel 2: charge source terms via V_WMMA_F32_16X16X4_F32.
//   phi_qq[j,ch] = sum_i (nc*g_ij)        * q[i,ch]
//   E_c [j,ch,c] = sum_i (-nc*D1_ij*d_c)  * q[i,ch]
// One wave (32 lanes) per 16-target tile; K looped 4 sources at a time.
// A layout (16x4 f32): lane%16 = M row; lanes<16 hold K=k0,k0+1 in v0,v1;
// lanes>=16 hold K=k0+2,k0+3.  B mirrors with N=lane%16.  D: VGPR v holds
// row v (lanes 0-15) / v+8 (lanes 16-31), col = lane%16.
// ---------------------------------------------------------------------------
__global__ __launch_bounds__(32) void ewald_charge(
    const float* __restrict__ qv, const float* __restrict__ r,
    float* __restrict__ ws) {
  const int lane = threadIdx.x;
  const int m    = lane & 15;   // A row / B-D column
  const int kh   = lane >> 4;   // K half
  const int jt   = blockIdx.x;  // target tile 0..31
  const int j    = jt * 16 + m;

  const float rjx = r[3 * j + 0];
  const float rjy = r[3 * j + 1];
  const float rjz = r[3 * j + 2];

  v8f accg = {0.f, 0.f, 0.f, 0.f, 0.f, 0.f, 0.f, 0.f};
  v8f accx = {0.f, 0.f, 0.f, 0.f, 0.f, 0.f, 0.f, 0.f};
  v8f accy = {0.f, 0.f, 0.f, 0.f, 0.f, 0.f, 0.f, 0.f};
  v8f accz = {0.f, 0.f, 0.f, 0.f, 0.f, 0.f, 0.f, 0.f};

  for (int k0 = 0; k0 < NPART; k0 += 4) {
    const int ib = k0 + kh * 2;
    v2f Ag, Ax, Ay, Az, Bq;
#pragma unroll
    for (int t = 0; t < 2; ++t) {
      const int i = ib + t;
      const float dx = rjx - r[3 * i + 0];
      const float dy = rjy - r[3 * i + 1];
      const float dz = rjz - r[3 * i + 2];
      float r2 = dx * dx + dy * dy + dz * dz;
      const float mask = (i == j) ? 0.f : 1.f;   // remove self-interaction
      r2 = (i == j) ? 1.f : r2;                  // safe diagonal
      const float rr  = sqrtf(r2);
      const float ir  = 1.0f / rr;
      const float ir2 = ir * ir, ir3 = ir2 * ir;
      const float E = erff(A_ERF * rr);
      const float G = GPREF * expf(-A2C * r2);
      const float kg = NCC * E * ir * mask;              // nc * g
      const float kd = -NCC * (G * ir2 - E * ir3) * mask;  // -nc * D1
      Ag[t] = kg;
      Ax[t] = kd * dx;
      Ay[t] = kd * dy;
      Az[t] = kd * dz;
      Bq[t] = (m < NQC) ? qv[i * NQC + m] : 0.f;  // cols 8..15 padded
    }
    // 8-arg form: (neg_a, A, neg_b, B, c_mod, C, reuse_a, reuse_b)
    accg = __builtin_amdgcn_wmma_f32_16x16x4_f32(false, Ag, false, Bq,
                                                 (short)0, accg, false, false);
    accx = __builtin_amdgcn_wmma_f32_16x16x4_f32(false, Ax, false, Bq,
                                                 (short)0, accx, false, false);
    accy = __builtin_amdgcn_wmma_f32_16x16x4_f32(false, Ay, false, Bq,
                                                 (short)0, accy, false, false);
    accz = __builtin_amdgcn_wmma_f32_16x16x4_f32(false, Az, false, Bq,
                                                 (short)0, accz, false, false);
  }

  if (m < NQC) {
#pragma unroll
    for (int v = 0; v < 8; ++v) {
      const int row = v + (kh ? 8 : 0);
      const int idx = (jt * 16 + row) * NQC + m;
      ws[W_PG * NJQ + idx]  = accg[v];
      ws[W_ECX * NJQ + idx] = accx[v];
      ws[W_ECY * NJQ + idx] = accy[v];
      ws[W_ECZ * NJQ + idx] = accz[v];
    }
  }
}

// ---------------------------------------------------------------------------
// Kernel 3: assemble induced charges/dipoles, outputs, and reduce pot.
// Single block, deterministic LDS tree reduction.
// ---------------------------------------------------------------------------
__global__ __launch_bounds__(512) void ewald_finalize(
    const float* __restrict__ qv, const float* __restrict__ u,
    const float* __restrict__ kappa, const float* __restrict__ alpha,
    const float* __restrict__ ws, float* __restrict__ out) {
  __shared__ double red[512];
  const int tid = threadIdx.x;
  double pacc = 0.0;

  for (int idx = tid; idx < NJQ; idx += 512) {
    const float pg  = ws[W_PG * NJQ + idx];
    const float pu  = ws[W_PU * NJQ + idx];
    const float pQ  = ws[W_PQ * NJQ + idx];
    const float Eux = ws[W_EUX * NJQ + idx];
    const float Euy = ws[W_EUY * NJQ + idx];
    const float Euz = ws[W_EUZ * NJQ + idx];
    const float EQx = ws[W_EQX * NJQ + idx];
    const float EQy = ws[W_EQY * NJQ + idx];
    const float EQz = ws[W_EQZ * NJQ + idx];
    const float sQ  = ws[W_SQQ * NJQ + idx];
    const float Ecx = ws[W_ECX * NJQ + idx];
    const float Ecy = ws[W_ECY * NJQ + idx];
    const float Ecz = ws[W_ECZ * NJQ + idx];

    const float qq  = qv[idx];
    const float uxv = u[idx * 3 + 0];
    const float uyv = u[idx * 3 + 1];
    const float uzv = u[idx * 3 + 2];
    const float kap = kappa[idx];
    const float alp = alpha[idx];

    const float phi = pg + pu + pQ;               // total e_phi
    const float Ex  = Ecx + Eux + EQx;            // total e_field
    const float Ey  = Ecy + Euy + EQy;
    const float Ez  = Ecz + Euz + EQz;

    out[1 + idx] = -kap * phi;                    // q_induced
    out[1 + NJQ + idx * 3 + 0] = alp * Ex;        // u_induced
    out[1 + NJQ + idx * 3 + 1] = alp * Ey;
    out[1 + NJQ + idx * 3 + 2] = alp * Ez;

    float p = 0.5f * pg * qq            // 0.5 * phi_qq . q
            + pu * qq                   // phi_u . q
            + pQ * qq                   // phi_Q . q
            - 0.5f * (uxv * Eux + uyv * Euy + uzv * Euz)  // -0.5 u.E_u
            - (uxv * EQx + uyv * EQy + uzv * EQz)         // -u.E_Q
            + sQ                        // 0.125 quad : QQ
            - 0.5f * kap * phi * phi    // 0.5 e_phi . q_induced
            - 0.5f * alp * (Ex * Ex + Ey * Ey + Ez * Ez); // -0.5 E.u_induced
    pacc += (double)p;
  }

  red[tid] = pacc;
  __syncthreads();
  for (int s = 256; s > 0; s >>= 1) {
    if (tid < s) red[tid] += red[tid + s];
    __syncthreads();
  }
  if (tid == 0) out[0] = (float)red[0];
}

// ---------------------------------------------------------------------------
extern "C" void kernel_launch(void* const* d_in, const int* in_sizes, int n_in,
                              void* d_out, int out_size, void* d_ws,
                              size_t ws_size, hipStream_t stream) {
  (void)in_sizes; (void)n_in; (void)out_size; (void)ws_size;
  const float* q     = (const float*)d_in[0];  // [512,8]
  const float* r     = (const float*)d_in[1];  // [512,3]
  // d_in[2] = cell (zero -> realspace branch), d_in[3] = batch : unused
  const float* u     = (const float*)d_in[4];  // [512,8,3]
  const float* quad  = (const float*)d_in[5];  // [512,8,3,3]
  const float* kappa = (const float*)d_in[6];  // [512,8]
  const float* alpha = (const float*)d_in[7];  // [512,8]
  float* out = (float*)d_out;                  // [1 + 4096 + 12288]
  float* ws  = (float*)d_ws;                   // needs 13*4096 floats (208 KB)

  // K1 and K2 write disjoint ws regions; K3 consumes both (stream-ordered).
  ewald_multipole<<<dim3(NJQ / 256), dim3(256), 0, stream>>>(r, u, quad, ws);
  ewald_charge<<<dim3(NPART / 16), dim3(32), 0, stream>>>(q, r, ws);
  ewald_finalize<<<dim3(1), dim3(512), 0, stream>>>(q, u, kappa, alpha, ws, out);
}